// BaseRenderer_53437983097195
// MI455X (gfx1250) — compile-verified
//
#include <hip/hip_runtime.h>
#include <hip/hip_bf16.h>
#include <stdint.h>

// D = A(16x4,f32) * B(4x16,f32) + C(16x16,f32) per wave32
typedef __attribute__((ext_vector_type(2))) float v2f;
typedef __attribute__((ext_vector_type(8))) float v8f;

#define KOBJ 4
#define NRAY 8192
#define SS   64
#define HH   64
#define KS   256   // KOBJ * SS

__device__ __forceinline__ float sigmoidf(float x) {
    return 1.0f / (1.0f + __expf(-x));
}

__global__ __launch_bounds__(32)
void BaseRenderer_render_kernel(
    const float*   __restrict__ ray_o,  // [N,3]
    const float*   __restrict__ ray_d,  // [N,3]
    const float*   __restrict__ nearp,  // [K,N]
    const float*   __restrict__ farp,   // [K,N]
    const uint8_t* __restrict__ masks,  // [K,N] (bool8)
    const float*   __restrict__ W1,     // [K,3,H]
    const float*   __restrict__ b1,     // [K,H]
    const float*   __restrict__ Wocc,   // [K,H,1]
    const float*   __restrict__ bocc,   // [K,1]
    const float*   __restrict__ Wrgb,   // [K,H,3]
    const float*   __restrict__ brgb,   // [K,3]
    float*         __restrict__ out)    // [N,12]
{
    __shared__ float hstage[16 * HH];   // one 16-sample tile of hidden activations
    __shared__ float zarr[KS];
    __shared__ float ch[4][KS];         // 0=alpha, 1..3 = rgb
    __shared__ int   sidx[KS];

    const int n    = blockIdx.x;        // one ray per wave32 block
    const int lane = threadIdx.x;       // 0..31
    const int lm   = lane & 15;
    const bool hi  = lane >= 16;

    const float ox = ray_o[n*3+0], oy = ray_o[n*3+1], oz = ray_o[n*3+2];
    const float dx = ray_d[n*3+0], dy = ray_d[n*3+1], dz = ray_d[n*3+2];

    for (int k = 0; k < KOBJ; ++k) {
        const float nr = nearp[k*NRAY + n];
        const float fr = farp[k*NRAY + n];
        const float mk = masks[k*NRAY + n] ? 1.0f : 0.0f;
        const float* W1k = W1 + k*3*HH;
        const float* b1k = b1 + k*HH;

        // ---- hoisted layer-1 B operands: [W1 rows x,y,z ; b1] 4x16 per N-tile ----
        // lanes 0-15 hold K rows 0,1 (x,y); lanes 16-31 hold rows 2,3 (z,bias).
        v2f w1bv[4];
        #pragma unroll
        for (int nt = 0; nt < 4; ++nt) {
            const int c = nt*16 + lm;
            const float* p0 = hi ? (W1k + 2*HH + c) : (W1k + 0*HH + c);
            const float* p1 = hi ? (b1k + c)        : (W1k + 1*HH + c);
            w1bv[nt].x = *p0;
            w1bv[nt].y = *p1;
        }

        // ---- hoisted layer-2 B operands: Wcat = [Wocc | Wrgb] (64x4 padded to 64x16) ----
        // chunk j covers hidden rows 4j..4j+3; lane holds rows (hi?4j+2,4j+3:4j,4j+1),
        // column = lm (0..3 meaningful; padded columns forced to zero, clamped index).
        float wb0[16], wb1[16];
        {
            const int   c    = lm;
            const float cmul = (c < 4) ? 1.0f : 0.0f;           // zero padded columns
            const int   cc   = (c >= 1 && c < 4) ? (c - 1) : 0; // clamped rgb column
            const bool  useOcc = (c == 0);
            #pragma unroll
            for (int j = 0; j < 16; ++j) {
                const int r0 = 4*j + (hi ? 2 : 0);
                const int r1 = r0 + 1;
                const float* q0 = useOcc ? (Wocc + k*HH + r0) : (Wrgb + (k*HH + r0)*3 + cc);
                const float* q1 = useOcc ? (Wocc + k*HH + r1) : (Wrgb + (k*HH + r1)*3 + cc);
                wb0[j] = (*q0) * cmul;
                wb1[j] = (*q1) * cmul;
            }
        }
        // per-channel output bias selected once (branchless cndmask chain)
        const float biasv = (lm == 0) ? bocc[k]
                          : (lm == 1) ? brgb[k*3+0]
                          : (lm == 2) ? brgb[k*3+1]
                          :             brgb[k*3+2];

        for (int mt = 0; mt < 4; ++mt) {
            // ---- layer 1: h = relu([pts,1] @ [W1;b1]) via WMMA f32 16x16x4 ----
            const int   m  = mt*16 + lm;                  // sample index 0..63
            const float t  = (float)m * (1.0f/63.0f);
            const float zm = nr + (fr - nr) * t;
            v2f a1;
            a1.x = hi ? (oz + dz*zm) : (ox + dx*zm);      // K2:z | K0:x
            a1.y = hi ? 1.0f         : (oy + dy*zm);      // K3:1 | K1:y
            if (!hi) zarr[k*SS + m] = zm * mk;

            #pragma unroll
            for (int nt = 0; nt < 4; ++nt) {
                const int c = nt*16 + lm;                 // hidden column
                v8f h = {};
                h = __builtin_amdgcn_wmma_f32_16x16x4_f32(
                        false, a1, false, w1bv[nt], (short)0, h, false, false);
                #pragma unroll
                for (int r = 0; r < 8; ++r) {
                    const int srow = r + (hi ? 8 : 0);    // D layout: lanes 16-31 hold M=r+8
                    hstage[srow*HH + c] = fmaxf(h[r], 0.0f);   // relu: single v_max_num_f32
                }
            }
            __syncthreads();

            // ---- layer 2: out[16x4] = h[16x64] @ Wcat[64x4], 16 chained K=4 WMMAs ----
            const v2f* hp = (const v2f*)hstage;           // 8B-aligned pair loads
            const int  abase = (lm*HH) / 2 + (hi ? 1 : 0);
            v8f o = {};
            #pragma unroll
            for (int j = 0; j < 16; ++j) {
                v2f a2 = hp[abase + 2*j];                 // hstage[lm*64 + 4j + (hi?2:0)], +1
                v2f b2; b2.x = wb0[j]; b2.y = wb1[j];
                o = __builtin_amdgcn_wmma_f32_16x16x4_f32(
                        false, a2, false, b2, (short)0, o, false, false);
            }
            // channel = lm (0=occ,1..3=rgb), sample = mt*16 + r + (hi?8:0)
            if (lm < 4) {
                #pragma unroll
                for (int r = 0; r < 8; ++r) {
                    const int srow = mt*16 + r + (hi ? 8 : 0);
                    ch[lm][k*SS + srow] = sigmoidf(o[r] + biasv) * mk;
                }
            }
            __syncthreads();
        }
    }

    // ---- stable rank-sort of 256 samples by z (matches stable argsort) ----
    #pragma unroll
    for (int e = 0; e < 8; ++e) {
        const int   i  = e*32 + lane;
        const float zi = zarr[i];
        int rank = 0;
        for (int j = 0; j < KS; ++j) {
            const float zj = zarr[j];
            rank += (zj < zi || (zj == zi && j < i)) ? 1 : 0;
        }
        sidx[rank] = i;
    }
    __syncthreads();

    // ---- serial cumprod alpha compositing (lane 0) ----
    if (lane == 0) {
        float trans = 1.0f, acc = 0.0f, dsum = 0.0f;
        float cr = 0.0f, cg = 0.0f, cb = 0.0f;
        float inst[KOBJ] = {0.0f, 0.0f, 0.0f, 0.0f};
        for (int p = 0; p < KS; ++p) {
            const int   i = sidx[p];
            const float a = ch[0][i];
            const float w = a * trans;
            acc  += w;
            dsum += w * zarr[i];
            cr   += w * ch[1][i];
            cg   += w * ch[2][i];
            cb   += w * ch[3][i];
            inst[i >> 6] += w;                  // object id = i / S
            trans *= (1.0f - a + 1e-10f);
        }
        const float nrm   = sqrtf(dx*dx + dy*dy + dz*dz);
        const float depth = dsum / (1e-10f + acc) / nrm;
        float* po = out + n*12;
        po[0] = cr;  po[1] = cg;  po[2] = cb;
        po[3] = acc; po[4] = depth;
        po[5] = inst[0]; po[6] = inst[1]; po[7] = inst[2]; po[8] = inst[3];
        po[9]  = ox + dx*depth;
        po[10] = oy + dy*depth;
        po[11] = oz + dz*depth;
    }
}

extern "C" void kernel_launch(void* const* d_in, const int* in_sizes, int n_in,
                              void* d_out, int out_size, void* d_ws, size_t ws_size,
                              hipStream_t stream) {
    const float*   ray_o = (const float*)d_in[0];
    const float*   ray_d = (const float*)d_in[1];
    const float*   nearp = (const float*)d_in[2];
    const float*   farp  = (const float*)d_in[3];
    const uint8_t* masks = (const uint8_t*)d_in[4];
    const float*   W1    = (const float*)d_in[5];
    const float*   b1    = (const float*)d_in[6];
    const float*   Wocc  = (const float*)d_in[7];
    const float*   bocc  = (const float*)d_in[8];
    const float*   Wrgb  = (const float*)d_in[9];
    const float*   brgb  = (const float*)d_in[10];
    float* out = (float*)d_out;

    BaseRenderer_render_kernel<<<NRAY, 32, 0, stream>>>(
        ray_o, ray_d, nearp, farp, masks, W1, b1, Wocc, bocc, Wrgb, brgb, out);
}